// BiasWeightLayerPrime_43568148250742
// MI455X (gfx1250) — compile-verified
//
#include <hip/hip_runtime.h>
#include <stdint.h>

// Problem constants (fixed by the reference harness).
#define NPRIMES 168
#define NCOLS   524288
#define NROWS   64
#define TILE    512      // columns per block
#define TPB     128      // threads per block; 4 consecutive cols per thread
#define PIPE    4        // async global->LDS pipeline depth (rows in flight)

// Compile-time table of the 168 primes < 1000 plus magic reciprocals for
// exact division: for c < 2^19, p < 2^10, M = floor(2^40/p)+1 gives
// floor(c/p) == (c*M) >> 40 exactly (c * (M*p - 2^40) <= c*p < 2^30 < 2^40).
struct PrimeTab { unsigned p[NPRIMES]; unsigned long long M[NPRIMES]; };

static constexpr PrimeTab make_tab() {
  PrimeTab t{};
  int cnt = 0;
  for (unsigned i = 2; i < 1000 && cnt < NPRIMES; ++i) {
    bool prime = true;
    for (int j = 0; j < cnt; ++j) {
      if (i % t.p[j] == 0) { prime = false; break; }
    }
    if (prime) {
      t.p[cnt] = i;
      t.M[cnt] = (1ULL << 40) / i + 1ULL;
      ++cnt;
    }
  }
  return t;
}

__device__ constexpr PrimeTab PT = make_tab();

__global__ __launch_bounds__(TPB) void prime_mask_stream_kernel(
    const float* __restrict__ x,
    const float* __restrict__ kern,
    const float* __restrict__ bias,
    float* __restrict__ out)
{
  // Single shared object -> starts at LDS offset 0; low 32 bits of the
  // generic pointer are the LDS byte offset (aperture base has zero low bits).
  __shared__ __align__(16) float xtile[PIPE * TILE];   // 8 KB

  const int      tid     = threadIdx.x;
  const unsigned c0      = (unsigned)blockIdx.x * TILE + (unsigned)tid * 4u;
  const size_t   colbase = (size_t)c0;

  // ---------------- Phase 1: w/b for this thread's 4 columns ----------------
  float w0 = 0.f, w1 = 0.f, w2 = 0.f, w3 = 0.f;
  float b0 = 0.f, b1 = 0.f, b2 = 0.f, b3 = 0.f;
  const float* kr = kern;
  const float* br = bias;
  for (int i = 0; i < NPRIMES; ++i) {
    const unsigned           p = PT.p[i];
    const unsigned long long M = PT.M[i];
    const unsigned q  = (unsigned)(((unsigned long long)c0 * M) >> 40);
    unsigned m0 = c0 - q * p;                         // c0 mod p (exact)
    unsigned m1 = m0 + 1u; m1 = (m1 == p) ? 0u : m1;  // (c0+1) mod p
    unsigned m2 = m1 + 1u; m2 = (m2 == p) ? 0u : m2;
    unsigned m3 = m2 + 1u; m3 = (m3 == p) ? 0u : m3;
    w0 += kr[m0]; w1 += kr[m1]; w2 += kr[m2]; w3 += kr[m3];
    b0 += br[m0]; b1 += br[m1]; b2 += br[m2]; b3 += br[m3];
    kr += 1000; br += 1000;
  }

  // ---------------- Phase 2: stream rows through an async LDS pipeline ------
  const unsigned sbase = (unsigned)(uintptr_t)(void*)xtile; // LDS byte base
  const unsigned slot  = (unsigned)tid * 16u;               // bytes in buffer

  // Prologue: stage rows 0..PIPE-2.
  for (int r = 0; r < PIPE - 1; ++r) {
    const float* gp    = x + (size_t)r * NCOLS + colbase;
    const unsigned la  = sbase + (unsigned)(r & (PIPE - 1)) * (TILE * 4u) + slot;
    asm volatile("global_load_async_to_lds_b128 %0, %1, off"
                 :: "v"(la), "v"(gp) : "memory");
  }

  for (int r = 0; r < NROWS; ++r) {
    if (r + PIPE - 1 < NROWS) {
      const float* gp   = x + (size_t)(r + PIPE - 1) * NCOLS + colbase;
      const unsigned la = sbase
                        + (unsigned)((r + PIPE - 1) & (PIPE - 1)) * (TILE * 4u)
                        + slot;
      asm volatile("global_load_async_to_lds_b128 %0, %1, off"
                   :: "v"(la), "v"(gp) : "memory");
      // 4 in flight; in-order completion => row r has landed when cnt <= 3.
      asm volatile("s_wait_asynccnt 0x3" ::: "memory");
    } else {
      asm volatile("s_wait_asynccnt 0x0" ::: "memory");
    }

    // Warm L2 beyond the LDS pipeline depth.
    if (r + 8 < NROWS)
      __builtin_prefetch(x + (size_t)(r + 8) * NCOLS + colbase, 0, 1);

    const float4 xv =
        *(const float4*)&xtile[(r & (PIPE - 1)) * TILE + tid * 4];

    float4 y;
    y.x = fmaf(xv.x, w0, b0);
    y.y = fmaf(xv.y, w1, b1);
    y.z = fmaf(xv.z, w2, b2);
    y.w = fmaf(xv.w, w3, b3);

    *(float4*)(out + (size_t)r * NCOLS + colbase) = y;
  }
}

extern "C" void kernel_launch(void* const* d_in, const int* in_sizes, int n_in,
                              void* d_out, int out_size, void* d_ws, size_t ws_size,
                              hipStream_t stream) {
  (void)in_sizes; (void)n_in; (void)out_size; (void)d_ws; (void)ws_size;
  const float* x    = (const float*)d_in[0];   // (64, 524288) fp32
  const float* kern = (const float*)d_in[1];   // (168, 1000) fp32
  const float* bias = (const float*)d_in[2];   // (168, 1000) fp32
  float*       out  = (float*)d_out;           // (64, 524288) fp32

  dim3 grid(NCOLS / TILE);   // 1024 blocks
  dim3 block(TPB);           // 128 threads = 4 waves
  prime_mask_stream_kernel<<<grid, block, 0, stream>>>(x, kern, bias, out);
}